// DeepUnfoldingNet_84859963835082
// MI455X (gfx1250) — compile-verified
//
#include <hip/hip_runtime.h>

typedef float v2f __attribute__((ext_vector_type(2)));
typedef float v8f __attribute__((ext_vector_type(8)));

// ---- geometry ----
#define TY 8            // tile rows
#define TX 16           // tile cols
#define TP 128          // pixels per tile
#define HY 16           // halo rows (fs=9 -> +-4)
#define HX 24           // halo cols
#define HP 384          // halo pixels
#define CH 48
#define S  52           // dword stride of pixel-major f32 rows (16B aligned, conflict-free)
#define AS 84           // attention-row stride (16B aligned)
#define NVIEW 9
#define IMG 128
#define PIXN (IMG*IMG)

// D[48 x 16*ntiles] = W[48x48] * X  via 12 chained V_WMMA_F32_16X16X4_F32 (exact fp32).
// f32 A (16x4): lane row = lane&15; VGPR0/1 = K = 2*khalf + {0,1}  -> one b64 per step.
// f32 B (4x16): lane col = lane&15; same K striping               -> one b64 per step.
__device__ __forceinline__ void gemm48(const float* __restrict__ W,
                                       const float* __restrict__ X,
                                       float* __restrict__ D, int ntiles) {
    const int wv   = threadIdx.x >> 5;
    const int lane = threadIdx.x & 31;
    const int colm = lane & 15;
    const int kh2  = (lane >> 4) << 1;           // 0 or 2
    const int jobs = 3 * ntiles;
    for (int j = wv; j < jobs; j += 8) {
        int mt = j % 3, nt = j / 3;
        const float* wr = W + (mt * 16 + colm) * S + kh2;
        const float* xr = X + (nt * 16 + colm) * S + kh2;
        v8f c = {};
#pragma unroll
        for (int kb = 0; kb < CH; kb += 4) {
            v2f a = *(const v2f*)(wr + kb);
            v2f b = *(const v2f*)(xr + kb);
            c = __builtin_amdgcn_wmma_f32_16x16x4_f32(false, a, false, b,
                                                      (short)0, c, false, false);
        }
        // D 16x16 f32: lane col = N, VGPR r -> row M = r + 8*khalf
        float* dr = D + (nt * 16 + colm) * S + mt * 16 + ((lane >> 4) << 3);
#pragma unroll
        for (int r = 0; r < 8; r += 2) *(v2f*)(dr + r) = (v2f){c[r], c[r + 1]};
    }
}

__launch_bounds__(256)
__global__ void ctg_fused_kernel(const float* __restrict__ warped,   // [1,9,48,128,128]
                                 const float* __restrict__ A,        // [1,9,48,128,128]
                                 const float* __restrict__ Wt,
                                 const float* __restrict__ Wp,
                                 const float* __restrict__ Wg,
                                 const float* __restrict__ Ww,
                                 float* __restrict__ out) {
    __shared__ __attribute__((aligned(16))) float sW[4 * CH * S];   // 39.0 KB weights
    __shared__ __attribute__((aligned(16))) float sX[HP * S];       // 78.0 KB warped halo
    __shared__ __attribute__((aligned(16))) float sPG[HP * S];      // 78.0 KB phi, then g
    __shared__ __attribute__((aligned(16))) float sA[TP * S];       // 26.0 KB A tile, then attn out
    __shared__ __attribute__((aligned(16))) float sTh[TP * S];      // 26.0 KB theta
    __shared__ __attribute__((aligned(16))) float sAtt[TP * AS];    // 42.0 KB logits

    const int tid  = threadIdx.x;
    const int view = blockIdx.z;
    const int gx0  = blockIdx.x * TX;
    const int gy0  = blockIdx.y * TY;

    const float* Av = A      + (size_t)view * CH * PIXN;
    const float* Wv = warped + (size_t)view * CH * PIXN;
    float*       Ov = out    + (size_t)view * CH * PIXN;

    // ---- zero halo buffer (border zero-padding == unfold semantics) ----
    for (int i = tid; i < HP * S; i += 256) sX[i] = 0.f;
    __syncthreads();

    // ---- stage weights + A tile + warped halo (all fp32, pixel-major) ----
    for (int i = tid; i < 4 * CH * CH; i += 256) {
        int mat = i / (CH * CH), rem = i % (CH * CH);
        int r = rem / CH, c = rem % CH;
        const float* wm = (mat == 0) ? Wt : (mat == 1) ? Wp : (mat == 2) ? Wg : Ww;
        sW[mat * CH * S + r * S + c] = wm[r * CH + c];
    }
    for (int i = tid; i < CH * TP; i += 256) {
        int c = i >> 7, p = i & (TP - 1);
        int py = p >> 4, px = p & 15;
        sA[p * S + c] = Av[(size_t)c * PIXN + (gy0 + py) * IMG + (gx0 + px)];
    }
    for (int i = tid; i < CH * HP; i += 256) {
        int c = i / HP, hp = i % HP;
        int hy = hp / HX, hx = hp % HX;
        int gy = gy0 + hy - 4, gx = gx0 + hx - 4;
        if ((unsigned)gy < (unsigned)IMG && (unsigned)gx < (unsigned)IMG)
            sX[hp * S + c] = Wv[(size_t)c * PIXN + gy * IMG + gx];
    }
    __syncthreads();

    // ---- theta = Wt*A (tile), phi = Wp*warped (halo): exact fp32 WMMA ----
    gemm48(sW + 0 * CH * S, sA, sTh, TP / 16);
    gemm48(sW + 1 * CH * S, sX, sPG, HP / 16);
    __syncthreads();

    // ---- logits: 2 threads per pixel, k-range split 41/40 ----
    const int pix  = tid & (TP - 1);
    const int part = tid >> 7;
    const int hy   = pix >> 4, hx = pix & 15;    // halo base = (hy+ky, hx+kx)
    float th[CH];
    {
        const float4* tr = (const float4*)(sTh + pix * S);
#pragma unroll
        for (int i = 0; i < CH / 4; ++i) {
            float4 f = tr[i];
            th[4*i] = f.x; th[4*i+1] = f.y; th[4*i+2] = f.z; th[4*i+3] = f.w;
        }
    }
    {
        const int k0 = part ? 41 : 0;
        const int k1 = part ? 81 : 41;
        for (int k = k0; k < k1; ++k) {
            int ky = k / 9, kx = k - ky * 9;
            const float4* pr = (const float4*)(sPG + ((hy + ky) * HX + hx + kx) * S);
            float a0 = 0.f, a1 = 0.f, a2 = 0.f, a3 = 0.f;
#pragma unroll
            for (int i = 0; i < CH / 4; ++i) {
                float4 f = pr[i];
                a0 += th[4*i]   * f.x;  a1 += th[4*i+1] * f.y;
                a2 += th[4*i+2] * f.z;  a3 += th[4*i+3] * f.w;
            }
            sAtt[pix * AS + k] = (a0 + a1) + (a2 + a3);  // OOB neighbors -> 0 (zero pad)
        }
    }
    __syncthreads();                              // logits done, phi reads done

    // ---- g = Wg*warped, overwrites phi buffer ----
    gemm48(sW + 2 * CH * S, sX, sPG, HP / 16);
    __syncthreads();

    // ---- softmax (recomputed per thread) + aggregation, channel split 24/24 ----
    {
        const float* ar = sAtt + pix * AS;
        float m = -3.0e38f;
        for (int k = 0; k < 81; ++k) m = fmaxf(m, ar[k]);
        float ssum = 0.f;
        for (int k = 0; k < 81; ++k) ssum += __expf(ar[k] - m);
        float inv = 1.f / ssum;

        const int cb = part * 24;
        float o[24];
#pragma unroll
        for (int i = 0; i < 24; ++i) o[i] = 0.f;
        for (int k = 0; k < 81; ++k) {
            float a = __expf(ar[k] - m) * inv;
            int ky = k / 9, kx = k - ky * 9;
            const float4* gr = (const float4*)(sPG + ((hy + ky) * HX + hx + kx) * S + cb);
#pragma unroll
            for (int i = 0; i < 6; ++i) {
                float4 f = gr[i];
                o[4*i]   += a * f.x;  o[4*i+1] += a * f.y;
                o[4*i+2] += a * f.z;  o[4*i+3] += a * f.w;
            }
        }
        float4* orow = (float4*)(sA + pix * S + cb);
#pragma unroll
        for (int i = 0; i < 6; ++i)
            orow[i] = make_float4(o[4*i], o[4*i+1], o[4*i+2], o[4*i+3]);
    }
    __syncthreads();

    // ---- final conv Ww*out + residual(A), fused epilogue ----
    {
        const int wv   = tid >> 5;
        const int lane = tid & 31;
        const int colm = lane & 15;
        const int kh2  = (lane >> 4) << 1;
        for (int j = wv; j < 3 * (TP / 16); j += 8) {
            int mt = j % 3, nt = j / 3;
            const float* wr = sW + 3 * CH * S + (mt * 16 + colm) * S + kh2;
            const float* xr = sA + (nt * 16 + colm) * S + kh2;
            v8f c = {};
#pragma unroll
            for (int kb = 0; kb < CH; kb += 4) {
                v2f a = *(const v2f*)(wr + kb);
                v2f b = *(const v2f*)(xr + kb);
                c = __builtin_amdgcn_wmma_f32_16x16x4_f32(false, a, false, b,
                                                          (short)0, c, false, false);
            }
            int p  = nt * 16 + colm;
            int py = p >> 4, px = p & 15;
            int gy = gy0 + py, gx = gx0 + px;
            int mb = mt * 16 + ((lane >> 4) << 3);
#pragma unroll
            for (int r = 0; r < 8; ++r) {
                size_t off = (size_t)(mb + r) * PIXN + gy * IMG + gx;
                Ov[off] = c[r] + Av[off];
            }
        }
    }
}

extern "C" void kernel_launch(void* const* d_in, const int* in_sizes, int n_in,
                              void* d_out, int out_size, void* d_ws, size_t ws_size,
                              hipStream_t stream) {
    const float* warped = (const float*)d_in[0];
    const float* A      = (const float*)d_in[1];
    const float* Wt     = (const float*)d_in[2];
    const float* Wp     = (const float*)d_in[3];
    const float* Wg     = (const float*)d_in[4];
    const float* Ww     = (const float*)d_in[5];
    float* outp = (float*)d_out;

    dim3 grid(IMG / TX, IMG / TY, NVIEW);   // 8 x 16 x 9
    dim3 block(256);                        // 8 wave32
    ctg_fused_kernel<<<grid, block, 0, stream>>>(warped, A, Wt, Wp, Wg, Ww, outp);
}